// DomainClassMixAugmentation_25297357373806
// MI455X (gfx1250) — compile-verified
//
#include <hip/hip_runtime.h>

// Problem constants (from reference): B=32, C=256, H=W=56
#define Bsz 32
#define Cch 256
#define HW  3136          // 56*56
#define HW4 784           // HW / 4 (float4 chunks)
#define BC  (Bsz * Cch)   // 8192 (b,c) rows

typedef float v4f __attribute__((ext_vector_type(4)));

// Low 32 bits of a flat LDS pointer == workgroup-relative LDS byte offset on gfx1250.
__device__ __forceinline__ unsigned lds_off(const void* p) {
    return (unsigned)(size_t)p;
}

// ---------------------------------------------------------------------------
// K1: per-(b,c) channel importance sums.
//   cim[bc] = sum_hw cg*x ; dim[bc] = sum_hw dg*x   (1/HW scale omitted:
//   strictly positive scaling preserves the quantile comparisons downstream)
//
// x is staged into lane-private LDS slots with CDNA5 async global->LDS b128
// copies, DOUBLE-BUFFERED: iteration i+1's DMA is issued before waiting on
// iteration i (s_wait_asynccnt 1; async loads complete in order). All async
// issues are branchless (tail indices clamped) so ASYNCcnt stays uniform
// across waves. Gradients stream with non-temporal hints so the 103MB x
// tensor stays resident in the 192MB L2 for the apply pass.
// ---------------------------------------------------------------------------
__global__ __launch_bounds__(256) void dcm_importance(
    const float* __restrict__ x, const float* __restrict__ cg,
    const float* __restrict__ dg, float* __restrict__ cim,
    float* __restrict__ dim_)
{
    __shared__ float xs[2][1024];  // double buffer: 256 lanes * float4 each
    __shared__ float rc[8], rd[8]; // 8 waves (wave32) per 256-thread block

    const int bc  = blockIdx.x;
    const int tid = threadIdx.x;
    const size_t row = (size_t)bc * HW;
    const float* xr = x  + row;
    const float* cr = cg + row;
    const float* dr = dg + row;

    const unsigned lb0 = lds_off(&xs[0][0]) + (unsigned)tid * 16u;
    const unsigned lb1 = lds_off(&xs[1][0]) + (unsigned)tid * 16u;

    float cacc = 0.f, dacc = 0.f;

    // Prologue: stage iteration 0 (all 256 lanes valid: 256 <= 784)
    {
        unsigned long long ga = (unsigned long long)(size_t)(xr + tid * 4);
        asm volatile("global_load_async_to_lds_b128 %0, %1, off"
                     :: "v"(lb0), "v"(ga) : "memory");
    }

#pragma unroll
    for (int it = 0; it < 4; ++it) {
        const int  raw    = it * 256 + tid;
        const bool active = raw < HW4;
        const int  v4i    = active ? raw : (HW4 - 1);  // clamped, branchless

        // Issue next iteration's async copy into the other buffer
        // (unconditional + clamped => ASYNCcnt uniform across all waves).
        if (it + 1 < 4) {
            int nraw = (it + 1) * 256 + tid;
            int nv   = nraw < HW4 ? nraw : (HW4 - 1);
            unsigned long long ga = (unsigned long long)(size_t)(xr + nv * 4);
            unsigned la = ((it + 1) & 1) ? lb1 : lb0;
            asm volatile("global_load_async_to_lds_b128 %0, %1, off"
                         :: "v"(la), "v"(ga) : "memory");
        }

        // Gradients: streamed, non-temporal (keep L2 for x); unconditional.
        v4f cv = __builtin_nontemporal_load((const v4f*)(cr + v4i * 4));
        v4f dv = __builtin_nontemporal_load((const v4f*)(dr + v4i * 4));

        // Wait for THIS iteration's x tile only (<=1 younger op outstanding).
        if (it < 3) asm volatile("s_wait_asynccnt 1" ::: "memory");
        else        asm volatile("s_wait_asynccnt 0" ::: "memory");

        if (active) {
            const float* src = (it & 1) ? &xs[1][tid * 4] : &xs[0][tid * 4];
            v4f xv = *(const v4f*)src;   // lane-private slot: no barrier needed
            cacc += cv.x * xv.x + cv.y * xv.y + cv.z * xv.z + cv.w * xv.w;
            dacc += dv.x * xv.x + dv.y * xv.y + dv.z * xv.z + dv.w * xv.w;
        }
    }

    // wave32 shuffle reduction, then cross-wave via LDS (single barrier)
    for (int off = 16; off > 0; off >>= 1) {
        cacc += __shfl_down(cacc, off, 32);
        dacc += __shfl_down(dacc, off, 32);
    }
    const int lane = tid & 31, wv = tid >> 5;
    if (lane == 0) { rc[wv] = cacc; rd[wv] = dacc; }
    __syncthreads();
    if (tid == 0) {
        float a = 0.f, b = 0.f;
#pragma unroll
        for (int i = 0; i < 8; ++i) { a += rc[i]; b += rd[i]; }
        cim[bc]  = a;
        dim_[bc] = b;
    }
}

// ---------------------------------------------------------------------------
// K2: per-sample quantile thresholds (bitonic sort of 256 values in LDS)
// and fold the four masks + mixup strengths into three per-(b,c) scalars:
//   w_own  = ds ? (cs ? s0 : s1) : 1          (self coefficient)
//   m_csds = cs & ds                           (gathered via same_idx, *(1-s0))
//   m_cgds = !cs & ds                          (gathered via diff_idx, *(1-s1))
// ---------------------------------------------------------------------------
__device__ __forceinline__ void bitonic256(float* s, int t) {
    for (int k = 2; k <= 256; k <<= 1) {
        for (int j = k >> 1; j > 0; j >>= 1) {
            __syncthreads();
            const int ixj = t ^ j;
            if (ixj > t) {
                const float a = s[t], b = s[ixj];
                const bool up = ((t & k) == 0);
                if ((a > b) == up) { s[t] = b; s[ixj] = a; }
            }
        }
    }
    __syncthreads();
}

__global__ __launch_bounds__(256) void dcm_thresholds(
    const float* __restrict__ cim, const float* __restrict__ dim_,
    const float* __restrict__ ms, float* __restrict__ w_own,
    float* __restrict__ m_csds, float* __restrict__ m_cgds)
{
    __shared__ float s[256];
    const int b = blockIdx.x;
    const int t = threadIdx.x;

    const float cv = cim[b * Cch + t];
    s[t] = cv;
    bitonic256(s, t);
    // median with linear interpolation: pos = 0.5*(256-1) = 127.5
    const float cthr = 0.5f * (s[127] + s[128]);
    __syncthreads();

    const float dv = dim_[b * Cch + t];
    s[t] = dv;
    bitonic256(s, t);
    // q = 0.8: pos = 0.8*255
    const float pos  = 0.8f * (float)(Cch - 1);
    const int   lo   = (int)pos;
    const float dthr = s[lo] + (pos - (float)lo) * (s[lo + 1] - s[lo]);
    __syncthreads();

    const bool cs = cv > cthr;
    const bool ds = dv > dthr;
    const float s0 = ms[b * 2 + 0];
    const float s1 = ms[b * 2 + 1];

    const int bc = b * Cch + t;
    w_own[bc]  = ds ? (cs ? s0 : s1) : 1.0f;
    m_csds[bc] = (cs && ds)  ? 1.0f : 0.0f;
    m_cgds[bc] = (!cs && ds) ? 1.0f : 0.0f;
}

// ---------------------------------------------------------------------------
// K3: apply. One block per (b,c): three uniform coefficients, then a 3-way
// fma over 784 float4s. x loads are regular-temporal (x is L2-resident from
// K1: 103MB < 192MB L2); output is stored non-temporal (never re-read).
// ---------------------------------------------------------------------------
__global__ __launch_bounds__(256) void dcm_apply(
    const float* __restrict__ x, const float* __restrict__ ms,
    const int* __restrict__ same_idx, const int* __restrict__ diff_idx,
    const float* __restrict__ w_own, const float* __restrict__ m_csds,
    const float* __restrict__ m_cgds, float* __restrict__ out)
{
    const int bc = blockIdx.x;
    const int b  = bc >> 8;
    const int c  = bc & (Cch - 1);

    const int sb = same_idx[b];
    const int db = diff_idx[b];
    const float s0 = ms[b * 2 + 0];
    const float s1 = ms[b * 2 + 1];

    const float c_self = w_own[bc];
    const float c_same = (1.0f - s0) * m_csds[sb * Cch + c];
    const float c_diff = (1.0f - s1) * m_cgds[db * Cch + c];

    const v4f* xa = (const v4f*)(x + (size_t)bc * HW);
    const v4f* xb = (const v4f*)(x + ((size_t)sb * Cch + c) * HW);
    const v4f* xc = (const v4f*)(x + ((size_t)db * Cch + c) * HW);
    v4f* o = (v4f*)(out + (size_t)bc * HW);

    for (int i = threadIdx.x; i < HW4; i += 256) {
        v4f r = c_self * xa[i] + c_same * xb[i] + c_diff * xc[i];
        __builtin_nontemporal_store(r, &o[i]);
    }
}

// ---------------------------------------------------------------------------
extern "C" void kernel_launch(void* const* d_in, const int* in_sizes, int n_in,
                              void* d_out, int out_size, void* d_ws, size_t ws_size,
                              hipStream_t stream) {
    const float* x  = (const float*)d_in[0];
    const float* cg = (const float*)d_in[1];
    const float* dg = (const float*)d_in[2];
    const float* ms = (const float*)d_in[3];
    // d_in[4]=y, d_in[5]=domain are only used on the host side of the reference
    const int* same_idx = (const int*)d_in[6];
    const int* diff_idx = (const int*)d_in[7];

    float* ws     = (float*)d_ws;   // 5 * 8192 floats = 160 KB of scratch
    float* cim    = ws + 0 * BC;
    float* dim_   = ws + 1 * BC;
    float* w_own  = ws + 2 * BC;
    float* m_csds = ws + 3 * BC;
    float* m_cgds = ws + 4 * BC;

    dcm_importance<<<BC, 256, 0, stream>>>(x, cg, dg, cim, dim_);
    dcm_thresholds<<<Bsz, 256, 0, stream>>>(cim, dim_, ms, w_own, m_csds, m_cgds);
    dcm_apply<<<BC, 256, 0, stream>>>(x, ms, same_idx, diff_idx,
                                      w_own, m_csds, m_cgds, (float*)d_out);
}